// IFNode_26663156974248
// MI455X (gfx1250) — compile-verified
//
#include <hip/hip_runtime.h>

// IFNode multi-step forward: h = v + x[t]; spike = (h >= 1); v = spike ? 0 : h
// Pure streaming scan over T=16 planes of N*F elements. HBM-bound:
// 256 MiB total traffic -> ~11.5 us floor at 23.3 TB/s. Strategy: b128
// non-temporal loads/stores, 16 planes buffered in registers per lane,
// serial scan entirely in VGPRs.

#define T_STEPS 16

// Native clang vector type: required by __builtin_nontemporal_load/store
// (HIP's float4 is a class and is rejected by the builtin).
typedef __attribute__((ext_vector_type(4))) float v4f;

__global__ __launch_bounds__(256) void ifnode_scan_v4(
    const v4f* __restrict__ x,   // [T][nf4] as 128-bit vectors
    v4f* __restrict__ out,       // [T][nf4] as 128-bit vectors
    int nf4)                     // (N*F)/4
{
    const int idx = blockIdx.x * blockDim.x + threadIdx.x;
    if (idx >= nf4) return;

    // Issue all 16 independent b128 loads up front (max MLP, NT hint:
    // single-use stream, do not pollute L2 with dead lines).
    v4f xv[T_STEPS];
#pragma unroll
    for (int t = 0; t < T_STEPS; ++t) {
        xv[t] = __builtin_nontemporal_load(&x[(size_t)t * (size_t)nf4 + (size_t)idx]);
    }

    float v0 = 0.0f, v1 = 0.0f, v2 = 0.0f, v3 = 0.0f;

#pragma unroll
    for (int t = 0; t < T_STEPS; ++t) {
        const float h0 = v0 + xv[t].x;
        const float h1 = v1 + xv[t].y;
        const float h2 = v2 + xv[t].z;
        const float h3 = v3 + xv[t].w;

        const bool f0 = (h0 >= 1.0f);
        const bool f1 = (h1 >= 1.0f);
        const bool f2 = (h2 >= 1.0f);
        const bool f3 = (h3 >= 1.0f);

        v4f sp;
        sp.x = f0 ? 1.0f : 0.0f;
        sp.y = f1 ? 1.0f : 0.0f;
        sp.z = f2 ? 1.0f : 0.0f;
        sp.w = f3 ? 1.0f : 0.0f;

        v0 = f0 ? 0.0f : h0;   // hard reset to v_reset = 0
        v1 = f1 ? 0.0f : h1;
        v2 = f2 ? 0.0f : h2;
        v3 = f3 ? 0.0f : h3;

        __builtin_nontemporal_store(sp, &out[(size_t)t * (size_t)nf4 + (size_t)idx]);
    }
}

// Scalar fallback for nf not divisible by 4 (not hit for the reference shape).
__global__ __launch_bounds__(256) void ifnode_scan_s(
    const float* __restrict__ x, float* __restrict__ out, int nf)
{
    const int idx = blockIdx.x * blockDim.x + threadIdx.x;
    if (idx >= nf) return;
    float v = 0.0f;
#pragma unroll
    for (int t = 0; t < T_STEPS; ++t) {
        const float h = v + __builtin_nontemporal_load(&x[(size_t)t * (size_t)nf + (size_t)idx]);
        const bool f = (h >= 1.0f);
        __builtin_nontemporal_store(f ? 1.0f : 0.0f,
                                    &out[(size_t)t * (size_t)nf + (size_t)idx]);
        v = f ? 0.0f : h;
    }
}

extern "C" void kernel_launch(void* const* d_in, const int* in_sizes, int n_in,
                              void* d_out, int out_size, void* d_ws, size_t ws_size,
                              hipStream_t stream)
{
    const float* x  = (const float*)d_in[0];
    float*      out = (float*)d_out;

    const int total = in_sizes[0];        // T * N * F
    const int nf    = total / T_STEPS;    // N * F (state plane)

    if ((nf & 3) == 0) {
        const int nf4 = nf >> 2;
        const int threads = 256;
        const int blocks  = (nf4 + threads - 1) / threads;
        ifnode_scan_v4<<<blocks, threads, 0, stream>>>(
            (const v4f*)x, (v4f*)out, nf4);
    } else {
        const int threads = 256;
        const int blocks  = (nf + threads - 1) / threads;
        ifnode_scan_s<<<blocks, threads, 0, stream>>>(x, out, nf);
    }
}